// ARMANet_56332791054868
// MI455X (gfx1250) — compile-verified
//
#include <hip/hip_runtime.h>
#include <hip/hip_bf16.h>

// ---------------------------------------------------------------------------
// ARMANet on MI455X (gfx1250, wave32).
//
// Pipeline (all fp32, matching the reference exactly):
//   1) deg[col] += 1  -> dinv = rsqrt(max(deg,1)) (0 where deg==0)
//   2) L1: [h|r] = x @ [W_init1|W_root1] (+b1 on root)  via V_WMMA_F32_16X16X4_F32
//   3) L1: agg[col] += dinv[row]*dinv[col] * h[row]     (wave-per-edge, lane=channel)
//   4) L1: out = relu(agg + r)    (relu(relu(.)) == relu(.))
//   5) L2: same with W_init2/W_root2/b2 (K=32)
//   6) fc: out[i] = h2[i,:] . W_fc + b_fc
//
// MI455X reasoning: per-layer feature table is 12.8 MB -> L2-resident (192 MB),
// so the scatter/gather phase is L2-atomic bound, not HBM bound. GEMMs are tiny
// (~2 GFLOP) but run on the fp32 WMMA pipe (16x16x4), which is exact.
// ---------------------------------------------------------------------------

typedef __attribute__((ext_vector_type(2))) float v2f;
typedef __attribute__((ext_vector_type(8))) float v8f;

#if defined(__has_builtin)
#if __has_builtin(__builtin_amdgcn_wmma_f32_16x16x4_f32)
#define HAVE_WMMA_F32X4 1
#endif
#endif

// ----------------------------- norm ----------------------------------------

__global__ __launch_bounds__(256) void degree_kernel(const int* __restrict__ col,
                                                     float* __restrict__ deg, int E) {
    int e = blockIdx.x * blockDim.x + threadIdx.x;
    if (e < E) atomicAdd(&deg[col[e]], 1.0f);
}

__global__ __launch_bounds__(256) void dinv_kernel(float* __restrict__ deg, int n) {
    int i = blockIdx.x * blockDim.x + threadIdx.x;
    if (i < n) {
        float d = deg[i];
        deg[i] = (d > 0.0f) ? rsqrtf(fmaxf(d, 1.0f)) : 0.0f;
    }
}

// --------------------- fused init/root GEMM (WMMA fp32) ---------------------
// Block = 128 threads = 4 waves. Wave w computes a 16x16 tile of the 64-wide
// concatenated output [init(0..31) | root(32..63)] for 16 rows.
// A (16x4 f32) lane layout per ISA: lanes 0-15 -> M=lane, K = k + {0,1};
//                                   lanes 16-31 -> M=lane-16, K = k + {2,3}.
// C/D layout: lane in [0,15]: N = lane, rows M = v (+8 for lanes 16-31).

template <int K>
__global__ __launch_bounds__(128) void gemm_init_root(
    const float* __restrict__ x,    // [nrows x K]
    const float* __restrict__ Wi,   // [K x 32]
    const float* __restrict__ Wr,   // [K x 32]
    const float* __restrict__ bias, // [32] (root only)
    float* __restrict__ h,          // [nrows x 32]  = x@Wi
    float* __restrict__ r,          // [nrows x 32]  = x@Wr + b
    int nrows) {
    const int wave = threadIdx.x >> 5; // 0..3 -> output col tile
    const int lane = threadIdx.x & 31;
    const int m0 = blockIdx.x * 16;
    const int n0 = wave * 16;

    const int arow = min(m0 + (lane & 15), nrows - 1); // clamp pad rows (discarded at store)
    const int akoff = (lane >> 4) * 2;                 // 0 or 2
    const int bcol = n0 + (lane & 15);                 // 0..63 (wave-uniform which half)
    const bool is_root = (n0 >= 32);
    const float* __restrict__ W = is_root ? Wr : Wi;
    const int wc = bcol & 31;

    v8f c = {};
#ifdef HAVE_WMMA_F32X4
    for (int k = 0; k < K; k += 4) {
        v2f a, b;
        a.x = x[arow * K + k + akoff];
        a.y = x[arow * K + k + akoff + 1];
        b.x = W[(k + akoff) * 32 + wc];
        b.y = W[(k + akoff + 1) * 32 + wc];
        // D = A(16x4) * B(4x16) + C ; exact fp32
        c = __builtin_amdgcn_wmma_f32_16x16x4_f32(false, a, false, b, (short)0, c,
                                                  false, false);
    }
#else
    // Scalar fallback (same result, no WMMA) — only if builtin missing.
    {
        const int rbase0 = m0 + (lane >> 4) * 8;
        for (int k = 0; k < K; ++k) {
            float bv = W[k * 32 + wc];
#pragma unroll
            for (int v = 0; v < 8; ++v) {
                int row = min(rbase0 + v, nrows - 1);
                c[v] += x[row * K + k] * bv;
            }
        }
    }
#endif

    const float badd = is_root ? bias[wc] : 0.0f;
    float* __restrict__ dst = is_root ? r : h;
    const int rbase = m0 + (lane >> 4) * 8;
#pragma unroll
    for (int v = 0; v < 8; ++v) {
        int row = rbase + v;
        if (row < nrows) dst[row * 32 + wc] = c[v] + badd;
    }
}

// ----------------- edge gather * norm -> scatter-add (L2 resident) ----------
// One wave per edge, lane = channel: 128B coalesced gather + coalesced atomics.

__global__ __launch_bounds__(256) void gather_scatter(
    const int* __restrict__ row, const int* __restrict__ col,
    const float* __restrict__ dinv, const float* __restrict__ h,
    float* __restrict__ agg, int E) {
    long long t = (long long)blockIdx.x * blockDim.x + threadIdx.x;
    int e = (int)(t >> 5);
    int c = (int)(t & 31);
    if (e >= E) return;
    int rs = row[e]; // wave-uniform (broadcast load)
    int cd = col[e];
    float nrm = dinv[rs] * dinv[cd];
    atomicAdd(&agg[cd * 32 + c], nrm * h[rs * 32 + c]);
}

// out = relu(agg + r) in place over agg
__global__ __launch_bounds__(256) void combine_relu(const float* __restrict__ r,
                                                    float* __restrict__ agg, int n) {
    int i = blockIdx.x * blockDim.x + threadIdx.x;
    if (i < n) agg[i] = fmaxf(agg[i] + r[i], 0.0f);
}

// out[i] = h[i,:] . Wfc + bfc
__global__ __launch_bounds__(256) void fc_kernel(const float* __restrict__ h,
                                                 const float* __restrict__ Wfc,
                                                 const float* __restrict__ bfc,
                                                 float* __restrict__ out, int n) {
    int i = blockIdx.x * blockDim.x + threadIdx.x;
    if (i >= n) return;
    float acc = bfc[0];
#pragma unroll
    for (int c = 0; c < 32; ++c) acc += h[i * 32 + c] * Wfc[c];
    out[i] = acc;
}

// ---------------------------------------------------------------------------

extern "C" void kernel_launch(void* const* d_in, const int* in_sizes, int n_in,
                              void* d_out, int out_size, void* d_ws, size_t ws_size,
                              hipStream_t stream) {
    const float* x   = (const float*)d_in[0];
    const int*   ei  = (const int*)d_in[1];
    const float* Wi1 = (const float*)d_in[2];
    const float* Wr1 = (const float*)d_in[3];
    const float* b1  = (const float*)d_in[4];
    const float* Wi2 = (const float*)d_in[5];
    const float* Wr2 = (const float*)d_in[6];
    const float* b2  = (const float*)d_in[7];
    const float* Wfc = (const float*)d_in[8];
    const float* bfc = (const float*)d_in[9];

    const int IN = 128, H = 32;
    const int N = in_sizes[0] / IN;     // 100000
    const int E = in_sizes[1] / 2;      // 1600000
    const int* erow = ei;
    const int* ecol = ei + E;

    // Workspace: dinv[N] | h[N*32] | r[N*32] | agg[N*32]  (~38.8 MB)
    float* dinv = (float*)d_ws;
    float* h    = dinv + N;
    float* rbuf = h + (size_t)N * H;
    float* agg  = rbuf + (size_t)N * H;

    const int NB256 = 256;
    const long long ethreads = (long long)E * 32;
    const int egrid = (int)((ethreads + NB256 - 1) / NB256);
    const int nH = N * H;

    // --- normalization ---
    hipMemsetAsync(dinv, 0, (size_t)N * sizeof(float), stream);
    degree_kernel<<<(E + NB256 - 1) / NB256, NB256, 0, stream>>>(ecol, dinv, E);
    dinv_kernel<<<(N + NB256 - 1) / NB256, NB256, 0, stream>>>(dinv, N);

    // --- layer 1 ---
    gemm_init_root<128><<<(N + 15) / 16, 128, 0, stream>>>(x, Wi1, Wr1, b1, h, rbuf, N);
    hipMemsetAsync(agg, 0, (size_t)nH * sizeof(float), stream);
    gather_scatter<<<egrid, NB256, 0, stream>>>(erow, ecol, dinv, h, agg, E);
    combine_relu<<<(nH + NB256 - 1) / NB256, NB256, 0, stream>>>(rbuf, agg, nH);
    // agg now holds layer-1 output

    // --- layer 2 ---
    gemm_init_root<32><<<(N + 15) / 16, 128, 0, stream>>>(agg, Wi2, Wr2, b2, h, rbuf, N);
    hipMemsetAsync(agg, 0, (size_t)nH * sizeof(float), stream);
    gather_scatter<<<egrid, NB256, 0, stream>>>(erow, ecol, dinv, h, agg, E);
    combine_relu<<<(nH + NB256 - 1) / NB256, NB256, 0, stream>>>(rbuf, agg, nH);

    // --- final linear ---
    fc_kernel<<<(N + NB256 - 1) / NB256, NB256, 0, stream>>>(agg, Wfc, bfc,
                                                             (float*)d_out, N);
}